// BoundaryLoss_558345749144
// MI455X (gfx1250) — compile-verified
//
#include <hip/hip_runtime.h>
#include <math.h>

typedef __attribute__((ext_vector_type(2))) float v2f;
typedef __attribute__((ext_vector_type(8))) float v8f;
typedef __attribute__((ext_vector_type(4))) unsigned int v4u;
typedef __attribute__((ext_vector_type(4))) int v4i;
typedef __attribute__((ext_vector_type(8))) int v8i;

#define BIGV 1.0e6f
#define HDIM 256
#define WDIM 256
#define NIMG 24          // B*C = 8*3
#define NPIX 65536       // 256*256
#define JT   32          // columns per pass-2 block

#if defined(__has_builtin)
#  if __has_builtin(__builtin_amdgcn_tensor_load_to_lds)
#    define HAVE_TDM 1
#  endif
#endif

// ---------------------------------------------------------------------------
// Kernel 1: 1-D EDT along W (reference pass 1), both polarities.
// grid = 48 blocks: blockIdx = m*2 + pol.  pol 0: zero_mask = (mask==0),
// pol 1: zero_mask = (mask!=0).  One wave per row via segmented shfl scans
// (exactly reproduces the reference's cummax formulation, incl. BIG
// sentinels).  pol==0 blocks also write counts[m] (empty-mask gate).
// ---------------------------------------------------------------------------
__global__ void edt_pass1(const int* __restrict__ target,
                          float* __restrict__ g2out,
                          int* __restrict__ counts) {
    const int m    = blockIdx.x >> 1;
    const int pol  = blockIdx.x & 1;
    const int t    = threadIdx.x;
    const int w    = t >> 5;
    const int lane = t & 31;
    const int mbase = m * NPIX;
    const int obase = (m * 2 + pol) * NPIX;

    __shared__ int sred[8];
    int cnt = 0;

    for (int rb = 0; rb < 32; ++rb) {
        const int r = w * 32 + rb;
        unsigned zbits = 0u;
        for (int s = 0; s < 8; ++s) {
            int v = target[mbase + r * WDIM + s * 32 + lane];
            cnt += (v != 0);
            bool zm = (pol == 0) ? (v == 0) : (v != 0);
            if (zm) zbits |= (1u << s);
        }
        // forward cummax of (zm ? col : -BIG)
        float lastv[8];
        float carry = -BIGV;
        for (int s = 0; s < 8; ++s) {
            float colf = (float)(s * 32 + lane);
            float x = ((zbits >> s) & 1u) ? colf : -BIGV;
            for (int off = 1; off < 32; off <<= 1) {
                float o = __shfl_up(x, off, 32);
                if (lane >= off) x = fmaxf(x, o);
            }
            x = fmaxf(x, carry);
            carry = __shfl(x, 31, 32);
            lastv[s] = x;
        }
        // reverse cummax of (zm ? -col : -BIG); nxt = -result
        carry = -BIGV;
        for (int s = 7; s >= 0; --s) {
            float colf = (float)(s * 32 + lane);
            float y = ((zbits >> s) & 1u) ? -colf : -BIGV;
            for (int off = 1; off < 32; off <<= 1) {
                float o = __shfl_down(y, off, 32);
                if (lane < 32 - off) y = fmaxf(y, o);
            }
            y = fmaxf(y, carry);
            carry = __shfl(y, 0, 32);
            float nxt = -y;
            float g = fminf(fminf(colf - lastv[s], nxt - colf), BIGV);
            g2out[obase + r * WDIM + s * 32 + lane] = g * g;
        }
    }

    if (pol == 0) {
        for (int off = 16; off >= 1; off >>= 1) cnt += __shfl_xor(cnt, off, 32);
        if (lane == 0) sred[w] = cnt;
        __syncthreads();
        if (t == 0) {
            int c = 0;
            for (int i = 0; i < 8; ++i) c += sred[i];
            counts[m] = c;
        }
    }
}

// ---------------------------------------------------------------------------
// Kernel 2: min-plus over H (reference pass 2), then sqrt, in place.
// The 32-column x 256-row g2 tile is pulled into LDS by the Tensor Data
// Mover: the D# uses pad_enable with pad_interval=4 (every 32 DWORDs) and
// pad_amount=0 (+1 DWORD) so the DMA itself produces the bank-conflict-free
// 33-float row stride the compute loop consumes.  One wave issues the TDM op
// (EXEC-ignored, SGPR descriptors), waits on TENSORcnt, block barrier
// publishes LDS.  (min,+) is a semiring WMMA can't express -> VALU inner
// loop with 4 running minima per thread (1 LDS read per 12 VALU).
// ---------------------------------------------------------------------------
__global__ void edt_pass2(float* __restrict__ g2buf) {
    __shared__ float tile[HDIM * (JT + 1)];
    const int b    = blockIdx.x;
    const int mp   = b >> 3;
    const int j0   = (b & 7) * JT;
    const int t    = threadIdx.x;
    const int base = mp * NPIX;

#if defined(HAVE_TDM)
    if (t < 32) {  // wave 0 issues the tensor DMA once
        unsigned lds_off = (unsigned)(unsigned long long)(void*)&tile[0];
        unsigned long long ga =
            (unsigned long long)(const void*)(g2buf + base + j0);
        v4u g0;
        g0[0] = 1u;                                   // count=1, user mode
        g0[1] = lds_off;                              // lds_addr
        g0[2] = (unsigned)(ga & 0xFFFFFFFFull);       // global_addr[31:0]
        g0[3] = (unsigned)((ga >> 32) & 0x01FFFFFFull) | (2u << 30); // type=2
        // data_size=4B(2<<16) | pad_enable(1<<20) | pad_interval=32dw(4<<22)
        // | pad_amount=1dw(0<<25); tensor_dim0=256, tensor_dim1=256,
        // tile_dim0=32, tile_dim1=256, tensor_dim0_stride=256
        v8i g1;
        g1[0] = 0x01120000;
        g1[1] = (int)(256u << 16);   // tensor_dim0 low16 in [31:16]
        g1[2] = (int)(256u << 16);   // tensor_dim0 hi=0 | tensor_dim1 low16
        g1[3] = (int)(32u << 16);    // tensor_dim1 hi=0 | tile_dim0=32
        g1[4] = 256;                 // tile_dim1=256, tile_dim2=0
        g1[5] = 256;                 // tensor_dim0_stride low32
        g1[6] = 0;
        g1[7] = 0;
        v4i gz = {0, 0, 0, 0};       // groups 2/3: tile_dim2/3=0 -> 2-D
#if __clang_major__ >= 23
        v8i gz8 = {0, 0, 0, 0, 0, 0, 0, 0};
        __builtin_amdgcn_tensor_load_to_lds(g0, g1, gz, gz, gz8, 0);
#else
        __builtin_amdgcn_tensor_load_to_lds(g0, g1, gz, gz, 0);
#endif
        __builtin_amdgcn_s_wait_tensorcnt(0);
    }
#else
    for (int it = 0; it < (HDIM * JT) / 256; ++it) {
        int idx = it * 256 + t;
        int k   = idx >> 5;
        int jj  = idx & 31;
        tile[k * (JT + 1) + jj] = g2buf[base + k * WDIM + j0 + jj];
    }
#endif
    __syncthreads();

    const int w  = t >> 5;   // i residue class (uniform per wave)
    const int jj = t & 31;   // column within tile (= lane)
    for (int c = 0; c < 8; ++c) {
        int   iu[4];
        float fi[4], d2[4];
        for (int u = 0; u < 4; ++u) {
            iu[u] = w + 8 * (c * 4 + u);
            fi[u] = (float)iu[u];
            d2[u] = 3.0e38f;
        }
        for (int k = 0; k < HDIM; ++k) {
            float g  = tile[k * (JT + 1) + jj];
            float kf = (float)k;
            for (int u = 0; u < 4; ++u) {
                float di = fi[u] - kf;
                float sq = di * di;           // separate mul+add to match jnp
                d2[u] = fminf(d2[u], sq + g);
            }
        }
        for (int u = 0; u < 4; ++u)
            g2buf[base + iu[u] * WDIM + j0 + jj] = sqrtf(d2[u]);
    }
}

// ---------------------------------------------------------------------------
// Kernel 3: loss accumulation via WMMA.  prod = sigmoid(pred) * d is fed as
// A into V_WMMA_F32_16X16X4_F32 with B = ones; every D entry is a row-sum of
// A, so (sum of all C regs over the wave)/16 == sum of products, independent
// of the A/D lane layout.  Sigmoid reciprocal via v_rcp_f32 (no IEEE divide).
// ---------------------------------------------------------------------------
__global__ void weighted_reduce(const float* __restrict__ pred,
                                const float* __restrict__ dbuf,
                                const int* __restrict__ counts,
                                float* __restrict__ partials) {
    const int b = blockIdx.x;        // 0..95
    const int m = b >> 2;
    const int q = b & 3;
    const int t = threadIdx.x;
    const int posb = (m * 2) * NPIX;
    const int negb = (m * 2 + 1) * NPIX;
    const float scale = (counts[m] == 0) ? 0.0f : 1.0f;

    v8f acc = {0.f, 0.f, 0.f, 0.f, 0.f, 0.f, 0.f, 0.f};
    v2f ones = {1.0f, 1.0f};
    float p0 = 0.0f;

    for (int it = 0; it < 64; ++it) {
        int   idx  = q * 16384 + it * 256 + t;
        float d    = (dbuf[posb + idx] - dbuf[negb + idx]) * scale;
        float x    = pred[m * NPIX + idx];
        float sg   = __builtin_amdgcn_rcpf(1.0f + __expf(-x));  // v_rcp_f32
        float prod = sg * d;
        if ((it & 1) == 0) {
            p0 = prod;
        } else {
            v2f a = {p0, prod};
            // D = A(16x4) * ones(4x16) + C   (accumulates across the loop)
            acc = __builtin_amdgcn_wmma_f32_16x16x4_f32(
                false, a, false, ones, (short)0, acc, false, false);
        }
    }

    float sL = acc[0] + acc[1] + acc[2] + acc[3] +
               acc[4] + acc[5] + acc[6] + acc[7];
    for (int off = 16; off >= 1; off >>= 1) sL += __shfl_xor(sL, off, 32);
    sL *= (1.0f / 16.0f);     // each row-sum appears 16x in D

    __shared__ float sred[8];
    if ((t & 31) == 0) sred[t >> 5] = sL;
    __syncthreads();
    if (t == 0) {
        float s = 0.f;
        for (int i = 0; i < 8; ++i) s += sred[i];
        partials[b] = s;
    }
}

// ---------------------------------------------------------------------------
// Kernel 4: deterministic final tree reduction of the 96 partials -> mean.
// ---------------------------------------------------------------------------
__global__ void finalize_loss(const float* __restrict__ partials,
                              float* __restrict__ out) {
    __shared__ float sred[128];
    int t = threadIdx.x;
    sred[t] = (t < 96) ? partials[t] : 0.0f;
    __syncthreads();
    for (int s = 64; s >= 1; s >>= 1) {
        if (t < s) sred[t] += sred[t + s];
        __syncthreads();
    }
    if (t == 0) out[0] = sred[0] / 1572864.0f;   // mean over 8*3*256*256
}

// ---------------------------------------------------------------------------
extern "C" void kernel_launch(void* const* d_in, const int* in_sizes, int n_in,
                              void* d_out, int out_size, void* d_ws, size_t ws_size,
                              hipStream_t stream) {
    (void)in_sizes; (void)n_in; (void)out_size; (void)ws_size;
    const float* pred   = (const float*)d_in[0];
    const int*   target = (const int*)d_in[1];

    // ws: g2/dist buffer (24*2*65536 f32 = 12.6 MB, sqrt'ed in place),
    // then counts (24 ints padded to 64), then 96 partial sums.
    char*  wsb      = (char*)d_ws;
    size_t g2bytes  = (size_t)NIMG * 2 * NPIX * sizeof(float);
    float* g2buf    = (float*)wsb;
    int*   counts   = (int*)(wsb + g2bytes);
    float* partials = (float*)(wsb + g2bytes + 64 * sizeof(int));
    float* outf     = (float*)d_out;

    edt_pass1<<<NIMG * 2, 256, 0, stream>>>(target, g2buf, counts);
    edt_pass2<<<NIMG * 2 * 8, 256, 0, stream>>>(g2buf);
    weighted_reduce<<<NIMG * 4, 256, 0, stream>>>(pred, g2buf, counts, partials);
    finalize_loss<<<1, 128, 0, stream>>>(partials, outf);
}